// CapsuleLayer_11347303596394
// MI455X (gfx1250) — compile-verified
//
#include <hip/hip_runtime.h>
#include <math.h>

// ---------------------------------------------------------------------------
// Capsule dynamic routing, CDNA5 (gfx1250), wave32.
//
// Key algebra #1: routing logits b_r[b,j,i] = dot( O_r[b,j,:], u_hat[b,j,i,:] )
// where O_r = sum of squash outputs of previous rounds (b starts at 0 and the
// update is linear in outputs). So we never materialize b or u_hat; each round
// recomputes u_hat via V_WMMA_F32_16X16X4_F32 with N = 16-batch tile.
//
// Key algebra #2: the weighted sum  s[d,b] += c_j(b) * (W x)[d,b]  is folded
// into the WMMA itself by pre-scaling the B operand lane-wise with c_j(b):
//   accS[j] = wmma(A, c_j * B, accS[j])
// so the accumulation pass has no transient tiles, no result moves and no
// scalar FMA accumulation.
//
// Register budget: sched_barriers split the j-loops into groups of 5 so the
// scheduler keeps at most ~5 transient tiles in flight (accS[10] = 80 VGPRs
// is the only large persistent state) -> allocation stays under 256 VGPRs.
//
// Round 0 has O == 0 -> c = 1/J exactly, so the logit pass is skipped
// (wave-uniform branch; EXEC stays all-ones for WMMA).
// ---------------------------------------------------------------------------

#define B_DIM   128
#define I_DIM   2048
#define J_DIM   10
#define DIN     8
#define DOUT    16
#define BT      16                  // batch tile (WMMA N)
#define NBT     (B_DIM / BT)        // 8
#define NCHUNK  16                  // i-chunks per batch tile
#define CHUNK_I (I_DIM / NCHUNK)    // 128
#define WAVES   4                   // waves per workgroup in S kernel
#define IPW     (CHUNK_I / WAVES)   // 32 i's per wave
#define PS_WG   (J_DIM * 8 * 32)    // 2560 floats of partial-s per workgroup

typedef float v2f __attribute__((ext_vector_type(2)));
typedef float v8f __attribute__((ext_vector_type(8)));

__global__ __launch_bounds__(256) void caps_zero(float* __restrict__ O, int n) {
  int t = blockIdx.x * blockDim.x + threadIdx.x;
  if (t < n) O[t] = 0.0f;
}

// S kernel: one workgroup = (batch tile, i-chunk). Each wave owns IPW i's.
__global__ __launch_bounds__(128)
void caps_route_partial(const float* __restrict__ x,   // [B, I, DIN]
                        const float* __restrict__ W,   // [J, I, DOUT, DIN]
                        const float* __restrict__ O,   // [B, J, DOUT]
                        float* __restrict__ ps,        // [NBT*NCHUNK][PS_WG]
                        int round)
{
  __shared__ float OLds[BT * J_DIM * DOUT];       // 10,240 B
  __shared__ float stage[WAVES][PS_WG];           // 40,960 B

  const int wg    = blockIdx.x;
  const int btile = wg / NCHUNK;
  const int chunk = wg % NCHUNK;
  const int b0    = btile * BT;
  const int tid   = threadIdx.x;
  const int lane  = tid & 31;
  const int wave  = tid >> 5;

  // Stage O tile into LDS (read-only this round). 2560 % 128 == 0: uniform.
  for (int e = tid; e < BT * J_DIM * DOUT; e += 128) {
    int bb = e / (J_DIM * DOUT);
    int r  = e - bb * (J_DIM * DOUT);
    OLds[e] = O[(size_t)(b0 + bb) * (J_DIM * DOUT) + r];
  }
  __syncthreads();

  // WMMA f32 16x16x4 operand mapping (ISA 7.12.2, 32-bit A 16x4 / B 4x16):
  //   lane&15 -> M (=d) for A, N (=batch) for B; lane>=16 selects K {2,3}.
  const int n    = lane & 15;
  const int hi   = lane >> 4;      // 0: lower half, 1: upper half
  const int koff = hi * 2;         // K offset within 4-wide chunk
  const int db   = hi * 8;         // d base held by this lane in C/D

  v8f accS[J_DIM];
  #pragma unroll
  for (int j = 0; j < J_DIM; ++j)
    accS[j] = (v8f){0.f, 0.f, 0.f, 0.f, 0.f, 0.f, 0.f, 0.f};

  const int i0 = chunk * CHUNK_I + wave * IPW;
  const float* xrow = x + ((size_t)(b0 + n) * I_DIM + i0) * DIN;
  const float* wrow = W + ((size_t)(i0)*DOUT + n) * DIN;  // + j*I*DOUT*DIN

  const v8f zc = (v8f){0.f, 0.f, 0.f, 0.f, 0.f, 0.f, 0.f, 0.f};

  for (int ii = 0; ii < IPW; ++ii) {
    // B operands: x[b0+n, i, k] ; shared across all j and both passes.
    const float* xp = xrow + (size_t)ii * DIN;
    v2f bOp0 = *(const v2f*)(xp + koff);       // k = koff, koff+1
    v2f bOp1 = *(const v2f*)(xp + 4 + koff);   // k = 4+koff, 5+koff
    const float* wp0 = wrow + (size_t)ii * (DOUT * DIN);

    float cj[J_DIM];

    if (round > 0) {
      // ---- Pass 1: logits (u_hat tile is transient).
      // Two groups of 5 with sched barriers to cap in-flight tiles.
      #pragma unroll
      for (int jg = 0; jg < 2; ++jg) {
        #pragma unroll
        for (int jl = 0; jl < J_DIM / 2; ++jl) {
          const int j = jg * (J_DIM / 2) + jl;
          const float* wp = wp0 + (size_t)j * I_DIM * DOUT * DIN;
          v2f a0 = *(const v2f*)(wp + koff);
          v2f a1 = *(const v2f*)(wp + 4 + koff);
          v8f c = __builtin_amdgcn_wmma_f32_16x16x4_f32(
              false, a0, false, bOp0, (short)0, zc, false, false);
          c = __builtin_amdgcn_wmma_f32_16x16x4_f32(
              false, a1, false, bOp1, (short)0, c, false, false);
          const float* Ob = &OLds[(n * J_DIM + j) * DOUT + db];
          float part = 0.f;
          #pragma unroll
          for (int p = 0; p < 8; ++p) part += Ob[p] * c[p];
          cj[j] = part + __shfl_xor(part, 16, 32);   // add other d-half
        }
        __builtin_amdgcn_sched_barrier(0);
      }
      // Softmax over j (per batch column; replicated across lane halves).
      float mx = cj[0];
      #pragma unroll
      for (int j = 1; j < J_DIM; ++j) mx = fmaxf(mx, cj[j]);
      float den = 0.f;
      #pragma unroll
      for (int j = 0; j < J_DIM; ++j) { cj[j] = expf(cj[j] - mx); den += cj[j]; }
      float inv = 1.0f / den;
      #pragma unroll
      for (int j = 0; j < J_DIM; ++j) cj[j] *= inv;
      __builtin_amdgcn_sched_barrier(0);
    } else {
      #pragma unroll
      for (int j = 0; j < J_DIM; ++j) cj[j] = 1.0f / (float)J_DIM;
    }

    // ---- Pass 2: accumulate s directly in the WMMA accumulator.
    // c_j(b) is folded into the B operand (lane holds batch column b).
    #pragma unroll
    for (int jg = 0; jg < 2; ++jg) {
      #pragma unroll
      for (int jl = 0; jl < J_DIM / 2; ++jl) {
        const int j = jg * (J_DIM / 2) + jl;
        const float* wp = wp0 + (size_t)j * I_DIM * DOUT * DIN;
        v2f a0 = *(const v2f*)(wp + koff);
        v2f a1 = *(const v2f*)(wp + 4 + koff);
        v2f bs0 = (v2f){cj[j] * bOp0.x, cj[j] * bOp0.y};
        v2f bs1 = (v2f){cj[j] * bOp1.x, cj[j] * bOp1.y};
        accS[j] = __builtin_amdgcn_wmma_f32_16x16x4_f32(
            false, a0, false, bs0, (short)0, accS[j], false, false);
        accS[j] = __builtin_amdgcn_wmma_f32_16x16x4_f32(
            false, a1, false, bs1, (short)0, accS[j], false, false);
      }
      __builtin_amdgcn_sched_barrier(0);
    }
  }

  // Fixed-order reduction across waves (deterministic, no fp atomics).
  #pragma unroll
  for (int j = 0; j < J_DIM; ++j)
    #pragma unroll
    for (int p = 0; p < 8; ++p)
      stage[wave][j * 256 + p * 32 + lane] = accS[j][p];
  __syncthreads();

  float* psw = ps + (size_t)wg * PS_WG;
  for (int e = tid; e < PS_WG; e += 128) {
    float ssum = stage[0][e];
    #pragma unroll
    for (int w = 1; w < WAVES; ++w) ssum += stage[w][e];
    psw[e] = ssum;
  }
}

// Q kernel: one thread per (b, j). Deterministic chunk sum, exact reference
// squash, writes out[b,j,:] and accumulates O += out for next round's logits.
__global__ __launch_bounds__(256)
void caps_squash(const float* __restrict__ ps,
                 float* __restrict__ O,
                 float* __restrict__ out)
{
  int t = blockIdx.x * blockDim.x + threadIdx.x;
  if (t >= B_DIM * J_DIM) return;
  const int b = t / J_DIM, j = t - b * J_DIM;
  const int btile = b >> 4, bl = b & 15;

  float s[DOUT];
  #pragma unroll
  for (int d = 0; d < DOUT; ++d) {
    const int p = d & 7, h = d >> 3;
    const float* base = ps + (size_t)(btile * NCHUNK) * PS_WG
                           + j * 256 + p * 32 + h * 16 + bl;
    float acc = 0.f;
    for (int ch = 0; ch < NCHUNK; ++ch) acc += base[(size_t)ch * PS_WG];
    s[d] = acc;
  }

  // squash(): scale from (v/5) norm, then layer-norm style normalization.
  float s2 = 0.f;
  #pragma unroll
  for (int d = 0; d < DOUT; ++d) { float v = s[d] * 0.2f; s2 += v * v; }
  float scale = 0.5f * s2 / (1.0f + 0.5f * s2) / sqrtf(s2 + 1e-7f);
  float mean = 0.f;
  #pragma unroll
  for (int d = 0; d < DOUT; ++d) mean += s[d];
  mean *= (1.0f / DOUT);
  float var = 0.f;
  #pragma unroll
  for (int d = 0; d < DOUT; ++d) { float v = s[d] - mean; var += v * v; }
  var *= (1.0f / DOUT);
  float rinv = 1.0f / sqrtf(var + 1e-6f);

  #pragma unroll
  for (int d = 0; d < DOUT; ++d) {
    float o = scale * (s[d] - mean) * rinv;
    out[(size_t)t * DOUT + d] = o;   // out[b,j,d]
    O[(size_t)t * DOUT + d] += o;    // logits accumulator for next round
  }
}

extern "C" void kernel_launch(void* const* d_in, const int* in_sizes, int n_in,
                              void* d_out, int out_size, void* d_ws, size_t ws_size,
                              hipStream_t stream) {
  (void)in_sizes; (void)n_in; (void)out_size; (void)ws_size;
  const float* x = (const float*)d_in[0];   // [128, 2048, 8]
  const float* W = (const float*)d_in[1];   // [10, 2048, 16, 8]
  float* out = (float*)d_out;               // [128, 10, 16]

  float* O  = (float*)d_ws;                       // 20480 floats
  float* ps = O + (size_t)B_DIM * J_DIM * DOUT;   // 128 * 2560 floats

  caps_zero<<<(B_DIM * J_DIM * DOUT + 255) / 256, 256, 0, stream>>>(
      O, B_DIM * J_DIM * DOUT);

  for (int r = 0; r < 3; ++r) {
    caps_route_partial<<<NBT * NCHUNK, 128, 0, stream>>>(x, W, O, ps, r);
    caps_squash<<<(B_DIM * J_DIM + 255) / 256, 256, 0, stream>>>(ps, O, out);
  }
}